// SuperGlueSinkhornKnoppLayer_88106959110214
// MI455X (gfx1250) — compile-verified
//
#include <hip/hip_runtime.h>

typedef __attribute__((ext_vector_type(2))) float v2f;
typedef __attribute__((ext_vector_type(4))) float v4f;
typedef __attribute__((ext_vector_type(8))) float v8f;

static constexpr int kB  = 8;
static constexpr int kN  = 1024;
static constexpr int kD  = 256;
static constexpr int kNP = 1025;   // N + dustbin
static constexpr int kLD = 1152;   // padded leading dim (9*128)
static constexpr int kChunks = 32; // row chunks per batch in pass kernel
static constexpr int kRows = 33;   // ceil(1025/32)
static constexpr float kLam  = 5.0f;
static constexpr float kClip = 3.0f; // 15/LAM

__device__ __forceinline__ float wave_reduce_sum(float v) {
#pragma unroll
  for (int off = 16; off > 0; off >>= 1) v += __shfl_xor(v, off, 32);
  return v;
}

// P0 entry: exp(-LAM * clip(-s, -3, 3)) == exp(LAM * clip(s, -3, 3))
__device__ __forceinline__ float pexp(float s) {
  float sc = fminf(fmaxf(s, -kClip), kClip);
  return __expf(kLam * sc);
}

// ---------------- L2 row normalization (one wave per row) ----------------
__global__ void __launch_bounds__(32)
k_normalize(const float* __restrict__ fA, const float* __restrict__ fB,
            float* __restrict__ nA, float* __restrict__ nB) {
  const int row = blockIdx.x;                      // 0..B*N-1
  const float* src = (blockIdx.y == 0) ? fA : fB;
  float* dst       = (blockIdx.y == 0) ? nA : nB;
  const int lane = threadIdx.x;
  const float* p = src + (size_t)row * kD;
  float v[kD / 32];
  float ss = 0.f;
#pragma unroll
  for (int k = 0; k < kD / 32; ++k) { v[k] = p[lane + 32 * k]; ss = fmaf(v[k], v[k], ss); }
  ss = wave_reduce_sum(ss);
  const float sc = rsqrtf(fmaxf(ss, 1e-12f));
  float* q = dst + (size_t)row * kD;
#pragma unroll
  for (int k = 0; k < kD / 32; ++k) q[lane + 32 * k] = v[k] * sc;
}

// ---------------- init r,c,y,z and per-batch Z base (dustbin cells) ------
__global__ void k_init(const float* __restrict__ dustbin,
                       float* __restrict__ r, float* __restrict__ c,
                       float* __restrict__ y, float* __restrict__ z,
                       float* __restrict__ Zs) {
  const int idx = blockIdx.x * blockDim.x + threadIdx.x;  // kB*kLD total
  if (idx >= kB * kLD) return;
  const int i = idx % kLD;
  const int b = idx / kLD;
  const float one = (i < kNP) ? 1.0f : 0.0f;  // zero padding lanes
  r[idx] = one; c[idx] = one;
  y[idx] = 0.f; z[idx] = 0.f;
  if (i == 0) Zs[b] = pexp(dustbin[0]) * (float)(2 * kNP - 1);
}

// ---------------- GEMM (f32 WMMA 16x16x4) + exp + Z partial --------------
// One wave computes a 16x64 tile: A-frag reused across 4 B-frags.
__global__ void __launch_bounds__(32)
k_gemm_exp(const float* __restrict__ A, const float* __restrict__ Bm,
           float* __restrict__ P0, float* __restrict__ Zs) {
  const int b  = blockIdx.z;
  const int m0 = blockIdx.y * 16;
  const int n0 = blockIdx.x * 64;
  const int lane = threadIdx.x;
  const int half = lane >> 4;    // 0: K pair {0,1}, 1: K pair {2,3}
  const int l16  = lane & 15;    // row (A) / col (B) within tile
  const float* Ab = A  + ((size_t)(b * kN + m0 + l16)) * kD + 2 * half;
  const float* Bb = Bm + ((size_t)(b * kN + n0 + l16)) * kD + 2 * half;
  v8f acc[4] = {};
  for (int kb = 0; kb < kD; kb += 4) {
    const v2f a  = *(const v2f*)(Ab + kb);
    const v2f b0 = *(const v2f*)(Bb + kb);
    const v2f b1 = *(const v2f*)(Bb + 16 * kD + kb);
    const v2f b2 = *(const v2f*)(Bb + 32 * kD + kb);
    const v2f b3 = *(const v2f*)(Bb + 48 * kD + kb);
    acc[0] = __builtin_amdgcn_wmma_f32_16x16x4_f32(false, a, false, b0, (short)0, acc[0], false, false);
    acc[1] = __builtin_amdgcn_wmma_f32_16x16x4_f32(false, a, false, b1, (short)0, acc[1], false, false);
    acc[2] = __builtin_amdgcn_wmma_f32_16x16x4_f32(false, a, false, b2, (short)0, acc[2], false, false);
    acc[3] = __builtin_amdgcn_wmma_f32_16x16x4_f32(false, a, false, b3, (short)0, acc[3], false, false);
  }
  float psum = 0.f;
#pragma unroll
  for (int t = 0; t < 4; ++t) {
#pragma unroll
    for (int e = 0; e < 8; ++e) {
      const float pv = pexp(acc[t][e]);
      const int row = m0 + half * 8 + e;       // C/D layout: M = vgpr (+8 upper half)
      const int col = n0 + 16 * t + l16;       // N = lane % 16
      P0[((size_t)(b * kNP + row)) * kLD + col] = pv;
      psum += pv;
    }
  }
  psum = wave_reduce_sum(psum);
  if (lane == 0) atomicAdd(&Zs[b], psum);
}

// ---------------- dustbin row/col + zero padding columns -----------------
__global__ void k_fill_pad(const float* __restrict__ dustbin, float* __restrict__ P0) {
  const int j = blockIdx.x * blockDim.x + threadIdx.x;
  const int i = blockIdx.y;
  const int b = blockIdx.z;
  if (j >= kLD) return;
  float* cell = &P0[((size_t)(b * kNP + i)) * kLD + j];
  if (j >= kNP) { *cell = 0.f; return; }        // padding -> 0 (no masking later)
  if (i == kN || j == kN) *cell = pexp(dustbin[0]);
}

// ---------------- fused Sinkhorn pass: y = P0*c, z = P0^T*r --------------
// One wave per (batch, 128-col strip, row chunk). Each lane owns 4 columns
// (b128 loads). z accumulates per-lane in registers; y via two overlapped
// wave shfl-reduction chains (rows unrolled x2) + 1 atomic per row.
__global__ void __launch_bounds__(32)
k_pass(const float* __restrict__ P0, const float* __restrict__ r,
       const float* __restrict__ c, float* __restrict__ y, float* __restrict__ z) {
  const int b  = blockIdx.z;
  const int j0 = blockIdx.x * 128 + threadIdx.x * 4;  // 0..1148 (padding = zeros)
  const int i0 = blockIdx.y * kRows;
  const int i1 = min(kNP, i0 + kRows);
  const float* Pb = P0 + (size_t)b * kNP * kLD;
  const v4f cj = *(const v4f*)(c + b * kLD + j0);
  v4f zacc = {0.f, 0.f, 0.f, 0.f};
  int i = i0;
  for (; i + 1 < i1; i += 2) {
    const v4f p0 = *(const v4f*)(Pb + (size_t)i * kLD + j0);
    const v4f p1 = *(const v4f*)(Pb + (size_t)(i + 1) * kLD + j0);
    const float r0 = r[b * kLD + i];
    const float r1 = r[b * kLD + i + 1];
    float y0 = fmaf(p0.x, cj.x, fmaf(p0.y, cj.y, fmaf(p0.z, cj.z, p0.w * cj.w)));
    float y1 = fmaf(p1.x, cj.x, fmaf(p1.y, cj.y, fmaf(p1.z, cj.z, p1.w * cj.w)));
#pragma unroll
    for (int off = 16; off > 0; off >>= 1) {   // two independent chains
      y0 += __shfl_xor(y0, off, 32);
      y1 += __shfl_xor(y1, off, 32);
    }
    if (threadIdx.x == 0) {
      atomicAdd(&y[b * kLD + i], y0);
      atomicAdd(&y[b * kLD + i + 1], y1);
    }
    zacc += p0 * r0;
    zacc += p1 * r1;
  }
  if (i < i1) {  // tail row
    const v4f p0 = *(const v4f*)(Pb + (size_t)i * kLD + j0);
    const float r0 = r[b * kLD + i];
    float y0 = fmaf(p0.x, cj.x, fmaf(p0.y, cj.y, fmaf(p0.z, cj.z, p0.w * cj.w)));
    y0 = wave_reduce_sum(y0);
    if (threadIdx.x == 0) atomicAdd(&y[b * kLD + i], y0);
    zacc += p0 * r0;
  }
  float* zp = z + b * kLD + j0;
  atomicAdd(zp + 0, zacc.x);
  atomicAdd(zp + 1, zacc.y);
  atomicAdd(zp + 2, zacc.z);
  atomicAdd(zp + 3, zacc.w);
}

// ---------------- scaling-vector update; re-zero y,z ---------------------
__global__ void k_update(float* __restrict__ r, float* __restrict__ c,
                         float* __restrict__ y, float* __restrict__ z,
                         const float* __restrict__ Zs) {
  const int idx = blockIdx.x * blockDim.x + threadIdx.x;  // kB*kLD total
  if (idx >= kB * kLD) return;
  const int i = idx % kLD;
  const int b = idx / kLD;
  const float yv = y[idx], zv = z[idx];
  if (i < kNP) {
    const float ab = (i == kN) ? (float)kN : 1.0f;
    const float Zb = Zs[b];
    r[idx] = Zb * ab / yv;
    c[idx] = Zb * ab / zv;
  } else {
    r[idx] = 0.f; c[idx] = 0.f;
  }
  y[idx] = 0.f; z[idx] = 0.f;
}

// ---------------- materialize final P = r_i * P0_ij * c_j / Z ------------
__global__ void k_finalize(const float* __restrict__ P0, const float* __restrict__ r,
                           const float* __restrict__ c, const float* __restrict__ Zs,
                           float* __restrict__ out) {
  const int j = blockIdx.x * blockDim.x + threadIdx.x;
  const int i = blockIdx.y;
  const int b = blockIdx.z;
  if (j >= kNP) return;
  const float invZ = 1.0f / Zs[b];
  out[((size_t)(b * kNP) + i) * kNP + j] =
      r[b * kLD + i] * P0[((size_t)(b * kNP + i)) * kLD + j] * c[b * kLD + j] * invZ;
}

extern "C" void kernel_launch(void* const* d_in, const int* in_sizes, int n_in,
                              void* d_out, int out_size, void* d_ws, size_t ws_size,
                              hipStream_t stream) {
  const float* fA = (const float*)d_in[0];
  const float* fB = (const float*)d_in[1];
  const float* dustbin = (const float*)d_in[2];
  float* out = (float*)d_out;

  float* ws = (float*)d_ws;
  float* nA = ws;                                   // B*N*D = 2,097,152
  float* nB = nA + (size_t)kB * kN * kD;            // 2,097,152
  float* P0 = nB + (size_t)kB * kN * kD;            // B*NP*LD = 9,446,400
  float* y  = P0 + (size_t)kB * kNP * kLD;          // B*LD
  float* z  = y + kB * kLD;
  float* r  = z + kB * kLD;
  float* c  = r + kB * kLD;
  float* Zs = c + kB * kLD;                         // B

  k_normalize<<<dim3(kB * kN, 2), 32, 0, stream>>>(fA, fB, nA, nB);
  k_init<<<(kB * kLD + 255) / 256, 256, 0, stream>>>(dustbin, r, c, y, z, Zs);
  k_gemm_exp<<<dim3(kN / 64, kN / 16, kB), 32, 0, stream>>>(nA, nB, P0, Zs);
  k_fill_pad<<<dim3((kLD + 255) / 256, kNP, kB), 256, 0, stream>>>(dustbin, P0);

  for (int it = 0; it < 100; ++it) {
    k_pass<<<dim3(kLD / 128, kChunks, kB), 32, 0, stream>>>(P0, r, c, y, z);
    k_update<<<(kB * kLD + 255) / 256, 256, 0, stream>>>(r, c, y, z, Zs);
  }

  k_finalize<<<dim3((kNP + 255) / 256, kNP, kB), 256, 0, stream>>>(P0, r, c, Zs, out);
}